// activation_module_48112223650434
// MI455X (gfx1250) — compile-verified
//
#include <hip/hip_runtime.h>

// Streaming SELU-axpy:
//   out[0:N)   = p                      (p = x[0:N))
//   out[N:2N)  = a * selu(p) + q        (q = x[N:2N))
// selu(v) = scale*v (v>0) else scale*alpha*expm1(v)   [JAX constants]
//
// Memory-bound: 20 B/element, 320 MB total @ 23.3 TB/s -> ~13.7 us floor.
// Working set > 192 MB L2 with zero reuse -> non-temporal hints everywhere.
// Branchless SELU: hardware v_exp_f32 for x<=-0.5, Horner FMA polynomial for
// (-0.5,0] (no cancellation), v_cndmask selects -> no EXEC divergence.

typedef __attribute__((ext_vector_type(4))) float f4;

#define SELU_SCALE       1.0507009873554805f
#define SELU_ALPHA_SCALE 1.7580993408473766f   // scale * alpha
#define LOG2E            1.4426950408889634f

// Branchless expm1 valid for x <= 0 (also benign for x > 0: unused there).
__device__ __forceinline__ float expm1_neg(float x) {
    // Path A (x <= -0.5): exp(x) - 1 via single hardware exp2.
    // No range reduction needed for negative inputs; deep underflow -> -1 exactly.
    float eA = __builtin_amdgcn_exp2f(x * LOG2E) - 1.0f;

    // Path B (-0.5 < x <= 0): Taylor/Horner, abs err ~1e-7 on the interval.
    float p = 1.9841270e-4f;                    // 1/5040
    p = fmaf(p, x, 1.3888889e-3f);              // 1/720
    p = fmaf(p, x, 8.3333333e-3f);              // 1/120
    p = fmaf(p, x, 4.1666667e-2f);              // 1/24
    p = fmaf(p, x, 1.6666667e-1f);              // 1/6
    p = fmaf(p, x, 0.5f);                       // 1/2
    p = fmaf(p, x, 1.0f);                       // 1
    float eB = x * p;                           // x + x^2/2 + ...

    return (x <= -0.5f) ? eA : eB;              // v_cndmask, no branch
}

__device__ __forceinline__ float selu_f(float v) {
    float neg = SELU_ALPHA_SCALE * expm1_neg(v);
    float pos = SELU_SCALE * v;
    return (v > 0.0f) ? pos : neg;              // v_cndmask, no branch
}

__global__ __launch_bounds__(256) void selu_axpy_v4_kernel(
    const f4* __restrict__ x,   // 2*N floats viewed as float4
    const f4* __restrict__ a,   // N floats viewed as float4
    f4* __restrict__ out,       // 2*N floats viewed as float4
    int n4)                     // N/4
{
    int i = blockIdx.x * blockDim.x + threadIdx.x;
    if (i >= n4) return;

    // Three NT streaming loads (global_load_b128 th:TH_LOAD_NT, clause'd)
    f4 p = __builtin_nontemporal_load(x + i);
    f4 q = __builtin_nontemporal_load(x + n4 + i);
    f4 s = __builtin_nontemporal_load(a + i);

    // Pass-through copy of p into the first half
    __builtin_nontemporal_store(p, out + i);

    f4 r;
#pragma unroll
    for (int j = 0; j < 4; ++j) {
        r[j] = fmaf(s[j], selu_f(p[j]), q[j]);
    }
    __builtin_nontemporal_store(r, out + n4 + i);
}

// Scalar tail for N % 4 != 0 (not hit for SIZE = 16M, kept for correctness).
__global__ void selu_axpy_tail_kernel(
    const float* __restrict__ x, const float* __restrict__ a,
    float* __restrict__ out, int n, int start)
{
    int i = start + blockIdx.x * blockDim.x + threadIdx.x;
    if (i >= n) return;
    float v = x[i];
    out[i]     = v;
    out[n + i] = fmaf(a[i], selu_f(v), x[n + i]);
}

extern "C" void kernel_launch(void* const* d_in, const int* in_sizes, int n_in,
                              void* d_out, int out_size, void* d_ws, size_t ws_size,
                              hipStream_t stream) {
    const float* x = (const float*)d_in[0];     // 2*N floats
    const float* a = (const float*)d_in[1];     // N floats
    float* out     = (float*)d_out;             // 2*N floats

    const int n  = in_sizes[1];                 // N = SIZE
    const int n4 = n >> 2;                      // float4 groups
    const int rem_start = n4 << 2;

    if (n4 > 0) {
        const int block = 256;
        const int grid  = (n4 + block - 1) / block;
        selu_axpy_v4_kernel<<<grid, block, 0, stream>>>(
            (const f4*)x, (const f4*)a, (f4*)out, n4);
    }
    if (rem_start < n) {
        const int rem   = n - rem_start;
        const int block = 64;
        const int grid  = (rem + block - 1) / block;
        selu_axpy_tail_kernel<<<grid, block, 0, stream>>>(x, a, out, n, rem_start);
    }
}